// GNA_841813590023
// MI455X (gfx1250) — compile-verified
//
#include <hip/hip_runtime.h>

#define D_ 128

typedef __attribute__((ext_vector_type(2))) float v2f;
typedef __attribute__((ext_vector_type(8))) float v8f;

// ---- order-preserving float <-> uint encoding for atomicMax on floats ----
__device__ __forceinline__ unsigned enc_f32(float f) {
  unsigned u = __float_as_uint(f);
  return (u & 0x80000000u) ? ~u : (u | 0x80000000u);
}
__device__ __forceinline__ float dec_f32(unsigned u) {
  return (u & 0x80000000u) ? __uint_as_float(u & 0x7FFFFFFFu)
                           : __uint_as_float(~u);
}
// enc(-inf) = ~0xFF800000 = 0x007FFFFF
#define ENC_NEG_INF 0x007FFFFFu

// Workgroup-relative LDS byte offset of a shared-memory pointer.
// Generic (flat) shared pointers carry the LDS offset in addr[31:0]
// (flat->LDS aperture mapping uses addr[31:0] directly).
__device__ __forceinline__ unsigned lds_off(void* p) {
  return (unsigned)(size_t)p;
}

// ---------------------------------------------------------------------------
// GEMM: dst[n,j] = sum_k A[n,k] * W[j,k] (+ bias[j] | + dst[n,j]) [relu]
// One block = 128 rows; 8 waves, each wave owns a 16-row stripe and walks the
// 8 column tiles. K-loop = 32 x V_WMMA_F32_16X16X4_F32 (full fp32 precision).
// W staged in LDS via GLOBAL_LOAD_ASYNC_TO_LDS_B128 (ASYNCcnt, no VGPR
// round-trip) with an XOR swizzle so B-fragment ds_load_b64 reads are
// bank-conflict-free; the async copy overlaps the A-fragment preload.
// ---------------------------------------------------------------------------
__global__ __launch_bounds__(256)
void gna_gemm_wmma(const float* __restrict__ A,
                   const float* __restrict__ W,
                   const float* __restrict__ bias,
                   float* __restrict__ dstp,
                   int N, int cFromDst, int doRelu)
{
  __shared__ float lW[D_ * D_];   // 64 KB, swizzled
  const int tid = threadIdx.x;
  const int rowBase = blockIdx.x * 128;

  // async stage W (128x128 f32) into LDS, XOR-swizzled on 4-float groups:
  // lds[j*128 + (k ^ ((j&31)<<2))]
#pragma unroll
  for (int i = 0; i < 16; ++i) {
    int idx4 = tid + i * 256;            // 0..4095 float4 index
    int j  = idx4 >> 5;                  // W row (0..127)
    int kg = (idx4 & 31) << 2;           // k group base
    int sw = kg ^ ((j & 31) << 2);
    unsigned dlds = lds_off(&lW[j * D_ + sw]);
    unsigned long long ga = (unsigned long long)(const void*)(W + (size_t)idx4 * 4);
    asm volatile("global_load_async_to_lds_b128 %0, %1, off"
                 :: "v"(dlds), "v"(ga) : "memory");
  }

  const int wave  = tid >> 5;
  const int lane  = tid & 31;
  const int mrow  = lane & 15;           // row within 16x16 tile
  const int khalf = lane >> 4;           // 0: K={0,1}, 1: K={2,3} of each slab

  // A-fragment preload (overlaps the async W copy): lane holds row
  // (rowBase + wave*16 + mrow), K pieces {4*kt + 2*khalf, +1}, kt=0..31.
  int arowg = rowBase + wave * 16 + mrow;
  if (arowg > N - 1) arowg = N - 1;      // clamp (OOB rows never stored)
  v2f areg[32];
#pragma unroll
  for (int kt = 0; kt < 32; ++kt)
    areg[kt] = *(const v2f*)&A[(size_t)arowg * D_ + kt * 4 + khalf * 2];

  asm volatile("s_wait_asynccnt 0" ::: "memory");
  __syncthreads();

  for (int j0 = 0; j0 < 8; ++j0) {
    const int col  = j0 * 16 + mrow;     // global output column (= W row)
    const int bswz = (col & 31) << 2;

    // C/D layout: VGPR r -> row (khalf*8 + r), col = lane&15
    v8f acc;
    if (cFromDst) {
#pragma unroll
      for (int r = 0; r < 8; ++r) {
        int row = rowBase + wave * 16 + khalf * 8 + r;
        acc[r] = (row < N) ? dstp[(size_t)row * D_ + col] : 0.0f;
      }
    } else {
      float bv = bias[col];
#pragma unroll
      for (int r = 0; r < 8; ++r) acc[r] = bv;
    }

#pragma unroll
    for (int kt = 0; kt < 32; ++kt) {
      int k = kt * 4 + khalf * 2;
      // B[k][n] = W[col][k]
      v2f bfrag = *(const v2f*)&lW[col * D_ + (k ^ bswz)];
      acc = __builtin_amdgcn_wmma_f32_16x16x4_f32(
                false, areg[kt], false, bfrag, (short)0, acc, false, false);
    }

#pragma unroll
    for (int r = 0; r < 8; ++r) {
      int row = rowBase + wave * 16 + khalf * 8 + r;
      if (row < N) {
        float v = acc[r];
        if (doRelu) v = fmaxf(v, 0.0f);
        dstp[(size_t)row * D_ + col] = v;
      }
    }
  }
}

// ---------------------------------------------------------------------------
// t[n] = dot(s2[n,:], a)  — one wave32 per node, float4 per lane + shfl tree
// ---------------------------------------------------------------------------
__global__ __launch_bounds__(256)
void gna_dot_a(const float* __restrict__ s2, const float* __restrict__ a,
               float* __restrict__ t, int N)
{
  int gw   = (blockIdx.x * blockDim.x + threadIdx.x) >> 5;
  int lane = threadIdx.x & 31;
  if (gw >= N) return;
  float4 x  = ((const float4*)s2)[(size_t)gw * 32 + lane];
  float4 av = ((const float4*)a)[lane];
  float p = x.x * av.x + x.y * av.y + x.z * av.z + x.w * av.w;
#pragma unroll
  for (int off = 16; off > 0; off >>= 1) p += __shfl_xor(p, off, 32);
  if (lane == 0) t[gw] = p;
}

// ---------------------------------------------------------------------------
// init: out <- broadcast(b1), m <- enc(-inf), z <- 0
// ---------------------------------------------------------------------------
__global__ void gna_init(float* __restrict__ ob, const float* __restrict__ b,
                         unsigned* __restrict__ m, float* __restrict__ z, int N)
{
  int i = blockIdx.x * blockDim.x + threadIdx.x;
  if (i < N * D_) ob[i] = b[i & (D_ - 1)];
  if (i < N) { m[i] = ENC_NEG_INF; z[i] = 0.0f; }
}

// ---------------------------------------------------------------------------
// segment-max of alpha = t[dst]-t[src] over dst (self loops appended)
// ---------------------------------------------------------------------------
__global__ void gna_edge_max(const long long* __restrict__ se,
                             const long long* __restrict__ de,
                             const float* __restrict__ t,
                             unsigned* __restrict__ m, int E, int N)
{
  int e = blockIdx.x * blockDim.x + threadIdx.x;
  if (e >= E + N) return;
  int si, di;
  if (e < E) { si = (int)se[e]; di = (int)de[e]; } else { si = di = e - E; }
  float alpha = t[di] - t[si];
  atomicMax(&m[di], enc_f32(alpha));
}

// segment-sum of exp(alpha - m[dst])
__global__ void gna_edge_sum(const long long* __restrict__ se,
                             const long long* __restrict__ de,
                             const float* __restrict__ t,
                             const unsigned* __restrict__ m,
                             float* __restrict__ z, int E, int N)
{
  int e = blockIdx.x * blockDim.x + threadIdx.x;
  if (e >= E + N) return;
  int si, di;
  if (e < E) { si = (int)se[e]; di = (int)de[e]; } else { si = di = e - E; }
  float alpha = t[di] - t[si];
  atomicAdd(&z[di], __expf(alpha - dec_f32(m[di])));
}

// ---------------------------------------------------------------------------
// out[dst,:] += softmax_w(e) * s2[src,:]  — one wave32 per edge,
// float4 gather per lane + 4 f32 atomic adds (L2-resident traffic)
// ---------------------------------------------------------------------------
__global__ __launch_bounds__(256)
void gna_edge_agg(const long long* __restrict__ se,
                  const long long* __restrict__ de,
                  const float* __restrict__ t,
                  const unsigned* __restrict__ m,
                  const float* __restrict__ z,
                  const float* __restrict__ s2,
                  float* __restrict__ ob, int E, int N)
{
  int gw   = (blockIdx.x * blockDim.x + threadIdx.x) >> 5;
  int lane = threadIdx.x & 31;
  if (gw >= E + N) return;
  int si, di;
  if (gw < E) { si = (int)se[gw]; di = (int)de[gw]; } else { si = di = gw - E; }
  float alpha = t[di] - t[si];
  float coef  = __expf(alpha - dec_f32(m[di])) / z[di];
  float4 v = ((const float4*)s2)[(size_t)si * 32 + lane];
  float* o = ob + (size_t)di * D_ + lane * 4;
  atomicAdd(o + 0, coef * v.x);
  atomicAdd(o + 1, coef * v.y);
  atomicAdd(o + 2, coef * v.z);
  atomicAdd(o + 3, coef * v.w);
}

// ---------------------------------------------------------------------------
extern "C" void kernel_launch(void* const* d_in, const int* in_sizes, int n_in,
                              void* d_out, int out_size, void* d_ws, size_t ws_size,
                              hipStream_t stream)
{
  const float*     s  = (const float*)d_in[0];
  const long long* ei = (const long long*)d_in[1];   // int64 edge_index
  const float*     w1 = (const float*)d_in[2];
  const float*     b1 = (const float*)d_in[3];
  const float*     w2 = (const float*)d_in[4];
  const float*     b2 = (const float*)d_in[5];
  const float*     a  = (const float*)d_in[6];

  const int N = in_sizes[0] / D_;      // 50000
  const int E = in_sizes[1] / 2;       // 800000
  const int L = in_sizes[3] / D_;      // 3

  // workspace layout (floats): bufA | bufB | s2 | t | m | z  (~77.4 MB)
  float* ws = (float*)d_ws;
  const size_t nd = (size_t)N * D_;
  float*    bufA = ws;
  float*    bufB = ws + nd;
  float*    s2b  = ws + 2 * nd;
  float*    tb   = ws + 3 * nd;
  unsigned* mb   = (unsigned*)(tb + N);
  float*    zb   = (float*)(mb + N);

  const long long* srcA = ei;
  const long long* dstA = ei + E;
  const int EN = E + N;

  dim3 blk(256);
  dim3 gGemm((N + 127) / 128);
  dim3 gDot(((unsigned)N * 32 + 255) / 256);
  dim3 gInit(((unsigned)N * D_ + 255) / 256);
  dim3 gEdge((EN + 255) / 256);
  dim3 gAgg(((unsigned)EN * 32 + 255) / 256);

  const float* cur = s;
  for (int l = 0; l < L; ++l) {
    const float* W1l = w1 + (size_t)l * D_ * D_;
    const float* b1l = b1 + (size_t)l * D_;
    const float* W2l = w2 + (size_t)l * D_ * D_;
    const float* b2l = b2 + (size_t)l * D_;
    const float* al  = a  + (size_t)l * D_;
    float* outl = (l == L - 1) ? (float*)d_out : ((l & 1) ? bufB : bufA);

    // s2 = cur @ W2^T + b2
    gna_gemm_wmma<<<gGemm, blk, 0, stream>>>(cur, W2l, b2l, s2b, N, 0, 0);
    // t = s2 @ a
    gna_dot_a<<<gDot, blk, 0, stream>>>(s2b, al, tb, N);
    // out <- b1, m <- -inf, z <- 0
    gna_init<<<gInit, blk, 0, stream>>>(outl, b1l, mb, zb, N);
    // segment softmax + weighted scatter
    gna_edge_max<<<gEdge, blk, 0, stream>>>(srcA, dstA, tb, mb, E, N);
    gna_edge_sum<<<gEdge, blk, 0, stream>>>(srcA, dstA, tb, mb, zb, E, N);
    gna_edge_agg<<<gAgg, blk, 0, stream>>>(srcA, dstA, tb, mb, zb, s2b, outl, E, N);
    // out = relu(cur @ W1^T + out)
    gna_gemm_wmma<<<gGemm, blk, 0, stream>>>(cur, W1l, nullptr, outl, N, 1, 1);
    cur = outl;
  }
}